// MeanToERA5_21534966022159
// MI455X (gfx1250) — compile-verified
//
#include <hip/hip_runtime.h>
#include <hip/hip_bf16.h>

// Segmented mean (MeanToERA5) for MI455X / gfx1250.
// x: [128][176400] fp32 channel planes, mapping: [176400] cluster ids (<10000),
// out: [128][10000] fp32 per-cluster means.
//
// Memory-bound (~96 MB => ~4.1 us @ 23.3 TB/s). Strategy:
//  * LDS-privatized scatter-add (ds_add_f32) into a 4ch x 10000 fp32
//    accumulator (160 KB of CDNA5's 320 KB WGP LDS) -> no hot-path global
//    atomics.
//  * mapping tiles staged into LDS by the Tensor Data Mover
//    (tensor_load_to_lds, TENSORcnt, double-buffered) where available.
//  * bulk x stream read as float4 (global_load_b128) + global_prefetch.
//  * cross-partial reduction + count-reciprocal scaling fused into a
//    v_wmma_f32_16x16x4_f32 ones-matrix reduction (S a template param so the
//    B-fragment loads are unconditional for the common S=8 case).

#define K_CLUSTERS 10000
#define N_CELLS    (420 * 420)      // 176400 (multiple of 16)
#define N_CH       128              // B*C = 8*16
#define CHG        4                // channels privatized per workgroup
#define N_GROUPS   (N_CH / CHG)     // 32
#define TILE       1024             // cells staged in LDS per iteration
#define BLOCK      256              // 8 wave32's

typedef float v2f __attribute__((ext_vector_type(2)));
typedef float v8f __attribute__((ext_vector_type(8)));

#if __has_builtin(__builtin_amdgcn_tensor_load_to_lds) && \
    __has_builtin(__builtin_amdgcn_s_wait_tensorcnt)
#define HAVE_TDM 1
typedef unsigned int u32x4 __attribute__((ext_vector_type(4)));
typedef int          i32x4 __attribute__((ext_vector_type(4)));
typedef int          i32x8 __attribute__((ext_vector_type(8)));

__device__ inline unsigned lds_off_of(const void* p) {
    // generic -> LDS addrspace; AS(3) pointers are 32-bit LDS byte offsets.
    return (unsigned)(unsigned long long)(
        (__attribute__((address_space(3))) const void*)p);
}

// 1-D TDM copy: elems 4-byte elements, global -> LDS (wave-level DMA).
__device__ inline void tdm_load_1d_to_lds(const void* gsrc, unsigned lds_off,
                                          int elems) {
    unsigned long long ga = (unsigned long long)gsrc;
    u32x4 g0;
    g0[0] = 1u;                                            // count=1 (valid)
    g0[1] = lds_off;                                       // lds_addr (bytes)
    g0[2] = (unsigned)(ga & 0xFFFFFFFFull);                // global_addr lo
    g0[3] = (unsigned)((ga >> 32) & 0x1FFFFFFull)          // global_addr hi
          | (2u << 30);                                    // type=2 (image)
    unsigned e = (unsigned)elems;
    i32x8 g1;
    g1[0] = (int)(2u << 16);                               // data_size=4B
    g1[1] = (int)((e & 0xFFFFu) << 16);                    // tensor_dim0 lo
    g1[2] = (int)(((e >> 16) & 0xFFFFu) | (1u << 16));     // dim0 hi | dim1=1
    g1[3] = (int)((e & 0xFFFFu) << 16);                    // tile_dim0
    g1[4] = 1;                                             // tile_dim1=1
    g1[5] = (int)e;                                        // dim0 stride
    g1[6] = 0;
    g1[7] = 0;
    i32x4 z4 = {0, 0, 0, 0};
#if __clang_major__ >= 23
    i32x8 z8 = {0, 0, 0, 0, 0, 0, 0, 0};
    __builtin_amdgcn_tensor_load_to_lds(g0, g1, z4, z4, z8, 0);
#else
    __builtin_amdgcn_tensor_load_to_lds(g0, g1, z4, z4, 0);
#endif
}
#else
#define HAVE_TDM 0
#endif

__global__ void zero_kernel(float* __restrict__ p, int n) {
    int i = blockIdx.x * blockDim.x + threadIdx.x;
    if (i < n) p[i] = 0.0f;
}

__global__ void counts_kernel(const int* __restrict__ mapping,
                              float* __restrict__ counts) {
    int i = blockIdx.x * blockDim.x + threadIdx.x;
    if (i < N_CELLS) {
        unsigned id = (unsigned)mapping[i];
        if (id < K_CLUSTERS) atomicAdd(&counts[id], 1.0f);
    }
}

// One workgroup = (cell-chunk s, channel-group g). Private LDS accumulator
// acc[CHG][K] (160 KB) + two staged id tiles (2 x 4 KB, double-buffered TDM).
__global__ void __launch_bounds__(BLOCK)
segsum_kernel(const float* __restrict__ x, const int* __restrict__ mapping,
              float* __restrict__ dst, int S, int cellsPerChunk, int atomicFlush) {
    extern __shared__ float smem[];
    float* acc  = smem;                                    // CHG*K floats
    int*   ids0 = (int*)(smem + CHG * K_CLUSTERS);         // TILE ints
    int*   ids1 = ids0 + TILE;                             // TILE ints
    int*   bufs[2] = {ids0, ids1};

    const int tid    = threadIdx.x;
    const int s      = blockIdx.x;
    const int g      = blockIdx.y;
    const int chbase = g * CHG;

    for (int i = tid; i < CHG * K_CLUSTERS; i += BLOCK) acc[i] = 0.0f;
    __syncthreads();

    const int cell0 = s * cellsPerChunk;                   // multiple of TILE
    const int cell1 = min(N_CELLS, cell0 + cellsPerChunk);
    const int nt    = (cell1 > cell0) ? (cell1 - cell0 + TILE - 1) / TILE : 0;

#if HAVE_TDM
    const unsigned idsOff0 = lds_off_of(ids0);
    const unsigned idsOff1 = lds_off_of(ids1);
    if (tid < 32 && nt > 0)                                // wave 0 issues DMA
        tdm_load_1d_to_lds(mapping + cell0, idsOff0, min(TILE, cell1 - cell0));
#endif

    for (int ti = 0; ti < nt; ++ti) {
        const int t   = cell0 + ti * TILE;
        const int cnt = min(TILE, cell1 - t);              // multiple of 4
        const int p   = ti & 1;
        const int* idbuf = bufs[p];

#if HAVE_TDM
        if (tid < 32) {
            if (ti + 1 < nt) {                             // prefetch next tile
                const int tn = t + TILE;
                tdm_load_1d_to_lds(mapping + tn, p ? idsOff0 : idsOff1,
                                   min(TILE, cell1 - tn));
                __builtin_amdgcn_s_wait_tensorcnt(1);      // current tile done
            } else {
                __builtin_amdgcn_s_wait_tensorcnt(0);
            }
        }
        __syncthreads();                                   // publish LDS tile
#else
        for (int i = 4 * tid; i < cnt; i += 4 * BLOCK)
            *(int4*)(bufs[p] + i) = *(const int4*)(mapping + t + i);
        __syncthreads();
#endif

        for (int i = 4 * tid; i < cnt; i += 4 * BLOCK) {
            const int4 id4 = *(const int4*)(idbuf + i);    // ds_load_b128
            const float* xp = x + (size_t)chbase * N_CELLS + (size_t)(t + i);
            __builtin_prefetch(xp + TILE);                 // global_prefetch_b8
#pragma unroll
            for (int j = 0; j < CHG; ++j) {
                const float4 v = *(const float4*)(xp + (size_t)j * N_CELLS);
                float* a = acc + j * K_CLUSTERS;
                if ((unsigned)id4.x < K_CLUSTERS) atomicAdd(&a[id4.x], v.x);
                if ((unsigned)id4.y < K_CLUSTERS) atomicAdd(&a[id4.y], v.y);
                if ((unsigned)id4.z < K_CLUSTERS) atomicAdd(&a[id4.z], v.z);
                if ((unsigned)id4.w < K_CLUSTERS) atomicAdd(&a[id4.w], v.w);
            }
        }
        __syncthreads();                                   // tile fully consumed
    }

    if (!atomicFlush) {
        // partials[(s*N_CH + ch)*K + k] — each slot written by exactly one WG.
#pragma unroll
        for (int j = 0; j < CHG; ++j) {
            float* p = dst + ((size_t)s * N_CH + chbase + j) * K_CLUSTERS;
            const float* a = acc + j * K_CLUSTERS;
            for (int k = 4 * tid; k < K_CLUSTERS; k += 4 * BLOCK)
                *(float4*)(p + k) = *(const float4*)(a + k);
        }
    } else {
        // Fallback: accumulate straight into d_out with global fp32 atomics.
#pragma unroll
        for (int j = 0; j < CHG; ++j) {
            float* p = dst + (size_t)(chbase + j) * K_CLUSTERS;
            for (int k = tid; k < K_CLUSTERS; k += BLOCK) {
                float v = acc[j * K_CLUSTERS + k];
                if (v != 0.0f) atomicAdd(&p[k], v);
            }
        }
    }
}

// Cross-partial reduction on the matrix unit: A = ones(16x4) makes
// D[m][col] = sum_k B[k][col] for every row m, independent of the exact
// B row->slot mapping. One wave produces 16 consecutive outputs; K=10000 is
// a multiple of 16, so a 16-wide group never crosses a channel row.
// S is a compile-time parameter: for S=8 the loads are unconditional
// (no exec-mask branches), odd S uses branchless clamped loads.
template <int S>
__global__ void __launch_bounds__(BLOCK)
reduce_wmma_kernel(const float* __restrict__ partials,
                   const float* __restrict__ counts,
                   float* __restrict__ out) {
    const int lane = threadIdx.x & 31;
    const int wave = blockIdx.x * (BLOCK / 32) + (threadIdx.x >> 5);
    const int e0   = wave * 16;
    if (e0 >= N_CH * K_CLUSTERS) return;                   // wave-uniform guard

    const int ch   = e0 / K_CLUSTERS;
    const int k0   = e0 - ch * K_CLUSTERS;
    const int col  = lane & 15;
    const int half = lane >> 4;
    const float* base = partials + (size_t)ch * K_CLUSTERS + k0 + col;

    v2f a; a[0] = 1.0f; a[1] = 1.0f;                       // 16x4 all-ones
    v8f c = {};                                            // 16x16 f32 accum

    constexpr int NPASS = (S + 3) / 4;
#pragma unroll
    for (int pass = 0; pass < NPASS; ++pass) {
        const int r0 = pass * 4 + 2 * half;                // rows r0, r0+1
        const int r1 = r0 + 1;
        const int cr0 = (r0 < S) ? r0 : (S - 1);           // clamped (branchless)
        const int cr1 = (r1 < S) ? r1 : (S - 1);
        float b0 = base[(size_t)cr0 * N_CH * K_CLUSTERS];
        float b1 = base[(size_t)cr1 * N_CH * K_CLUSTERS];
        v2f b;
        b[0] = (r0 < S) ? b0 : 0.0f;
        b[1] = (r1 < S) ? b1 : 0.0f;
        c = __builtin_amdgcn_wmma_f32_16x16x4_f32(
                /*neg_a=*/false, a, /*neg_b=*/false, b,
                /*c_mod=*/(short)0, c, /*reuse_a=*/false, /*reuse_b=*/false);
    }

    if (lane < 16) {                                       // D[0][col] in c[0]
        float cnt = counts[k0 + col];
        float inv = 1.0f / fmaxf(cnt, 1.0f);
        out[(size_t)ch * K_CLUSTERS + k0 + col] = c[0] * inv;
    }
}

__global__ void finalize_inplace_kernel(float* __restrict__ out,
                                        const float* __restrict__ counts) {
    int i = blockIdx.x * blockDim.x + threadIdx.x;
    if (i < N_CH * K_CLUSTERS) {
        int k = i % K_CLUSTERS;
        out[i] *= 1.0f / fmaxf(counts[k], 1.0f);
    }
}

extern "C" void kernel_launch(void* const* d_in, const int* in_sizes, int n_in,
                              void* d_out, int out_size, void* d_ws, size_t ws_size,
                              hipStream_t stream) {
    const float* x       = (const float*)d_in[0];
    const int*   mapping = (const int*)d_in[1];
    float*       out     = (float*)d_out;

    float* counts = (float*)d_ws;
    const size_t partialOff = (((size_t)K_CLUSTERS * 4) + 255) & ~(size_t)255;
    const size_t perS       = (size_t)N_CH * K_CLUSTERS * 4;

    int S = 0;
    if (ws_size > partialOff) S = (int)((ws_size - partialOff) / perS);
    if (S > 8) S = 8;

    zero_kernel<<<(K_CLUSTERS + 255) / 256, 256, 0, stream>>>(counts, K_CLUSTERS);
    counts_kernel<<<(N_CELLS + 255) / 256, 256, 0, stream>>>(mapping, counts);

    const size_t lds_bytes =
        (size_t)CHG * K_CLUSTERS * 4 + (size_t)2 * TILE * 4;   // 168192 B

    if (S >= 1) {
        float* partials = (float*)((char*)d_ws + partialOff);
        // chunk size rounded to TILE so every tile is 16B-aligned, mult of 4
        int cpc = ((N_CELLS + S - 1) / S + TILE - 1) / TILE * TILE;
        segsum_kernel<<<dim3(S, N_GROUPS), BLOCK, lds_bytes, stream>>>(
            x, mapping, partials, S, cpc, /*atomicFlush=*/0);
        const int nblocks = (N_CH * K_CLUSTERS) / (16 * (BLOCK / 32)); // 10000
        switch (S) {
        case 1: reduce_wmma_kernel<1><<<nblocks, BLOCK, 0, stream>>>(partials, counts, out); break;
        case 2: reduce_wmma_kernel<2><<<nblocks, BLOCK, 0, stream>>>(partials, counts, out); break;
        case 3: reduce_wmma_kernel<3><<<nblocks, BLOCK, 0, stream>>>(partials, counts, out); break;
        case 4: reduce_wmma_kernel<4><<<nblocks, BLOCK, 0, stream>>>(partials, counts, out); break;
        case 5: reduce_wmma_kernel<5><<<nblocks, BLOCK, 0, stream>>>(partials, counts, out); break;
        case 6: reduce_wmma_kernel<6><<<nblocks, BLOCK, 0, stream>>>(partials, counts, out); break;
        case 7: reduce_wmma_kernel<7><<<nblocks, BLOCK, 0, stream>>>(partials, counts, out); break;
        default: reduce_wmma_kernel<8><<<nblocks, BLOCK, 0, stream>>>(partials, counts, out); break;
        }
    } else {
        // Workspace too small: atomic flush directly into d_out.
        zero_kernel<<<(N_CH * K_CLUSTERS + 255) / 256, 256, 0, stream>>>(
            out, N_CH * K_CLUSTERS);
        const int SB = 8;
        int cpc = ((N_CELLS + SB - 1) / SB + TILE - 1) / TILE * TILE;
        segsum_kernel<<<dim3(SB, N_GROUPS), BLOCK, lds_bytes, stream>>>(
            x, mapping, out, SB, cpc, /*atomicFlush=*/1);
        finalize_inplace_kernel<<<(N_CH * K_CLUSTERS + 255) / 256, 256, 0, stream>>>(
            out, counts);
    }
}